// IndexPutVoxelizer_88914412961980
// MI455X (gfx1250) — compile-verified
//
#include <hip/hip_runtime.h>
#include <stdint.h>

#define VOXEL 32
#define FDIM 256
#define PTS_PER_WAVE 2
#define WAVES_PER_BLOCK 8
#define PTS_PER_BLOCK (PTS_PER_WAVE * WAVES_PER_BLOCK)  // 16 points, 16 KB LDS

// gfx1250 async global->LDS path (ASYNCcnt), guarded so compile never fails.
#if defined(__has_builtin)
#  if __has_builtin(__builtin_amdgcn_global_load_async_to_lds_b128) && \
      __has_builtin(__builtin_amdgcn_s_wait_asynccnt)
#    define USE_ASYNC_LDS 1
#  endif
#endif
#ifndef USE_ASYNC_LDS
#  define USE_ASYNC_LDS 0
#endif

#if USE_ASYNC_LDS
// Exact parameter types of __builtin_amdgcn_global_load_async_to_lds_b128:
// (v4i addrspace(1)*, v4i addrspace(3)*, imm offset, imm cpol)
typedef int v4i __attribute__((vector_size(16)));
typedef __attribute__((address_space(1))) v4i gl_v4i;
typedef __attribute__((address_space(3))) v4i ds_v4i;
#endif

// Order-preserving float -> u32 bijection: f>=0 -> bits|0x80000000, f<0 -> ~bits.
// Unsigned max on encodings == float max. Sentinel 0u (= encoded -NaN) marks
// "never touched" — unreachable from ordinary float inputs.
__device__ __forceinline__ unsigned enc_ordered(float f) {
    unsigned u = __float_as_uint(f);
    return (u & 0x80000000u) ? ~u : (u | 0x80000000u);
}

__global__ __launch_bounds__(256) void init_zero_kernel(uint4* __restrict__ out, int n4) {
    int i = blockIdx.x * blockDim.x + threadIdx.x;
    int stride = gridDim.x * blockDim.x;
    uint4 z; z.x = 0u; z.y = 0u; z.z = 0u; z.w = 0u;
    for (; i < n4; i += stride) out[i] = z;
}

__global__ __launch_bounds__(256) void scatter_max_kernel(
        const float* __restrict__ feat,    // [total_pts, FDIM]
        const float* __restrict__ coords,  // [total_pts, 2]
        unsigned* __restrict__ out,        // [B*V*V, FDIM] encoded u32
        int total_pts, int pts_per_batch)
{
#if USE_ASYNC_LDS
    __shared__ __align__(16) float lds[PTS_PER_BLOCK * FDIM];
#endif
    const int lane = threadIdx.x & 31;   // wave32
    const int wave = threadIdx.x >> 5;
    const int p0 = blockIdx.x * PTS_PER_BLOCK + wave * PTS_PER_WAVE;

#if USE_ASYNC_LDS
    float* wlds = lds + (size_t)(wave * PTS_PER_WAVE) * FDIM;
    for (int q = 0; q < PTS_PER_WAVE; ++q) {
        int p = p0 + q;
        if (p >= total_pts) break;
        const float* src = feat + (size_t)p * FDIM;
        float* dstl = wlds + q * FDIM;
        // 256 floats = 64 x 16B chunks; 32 lanes -> 2 async b128 instrs/point,
        // each instruction moves 512 B per wave directly into LDS.
        for (int j = 0; j < 2; ++j) {
            int c4 = (j * 32 + lane) * 4;
            __builtin_amdgcn_global_load_async_to_lds_b128(
                (gl_v4i*)(src + c4),
                (ds_v4i*)(dstl + c4),
                0, 0);
        }
    }
    __builtin_amdgcn_s_wait_asynccnt(0);   // s_wait_asynccnt 0
    __asm__ __volatile__("" ::: "memory");
#endif

    for (int q = 0; q < PTS_PER_WAVE; ++q) {
        int p = p0 + q;
        if (p >= total_pts) break;
        // Quantize coords (is_norm=False path: coords already in [0,1)).
        float cx = coords[2 * (size_t)p + 0];
        float cy = coords[2 * (size_t)p + 1];
        int gx = (int)(cx * (float)(VOXEL - 1));   // trunc like astype(int32)
        int gy = (int)(cy * (float)(VOXEL - 1));
        gx = min(max(gx, 0), VOXEL - 1);
        gy = min(max(gy, 0), VOXEL - 1);
        int cell = (p / pts_per_batch) * (VOXEL * VOXEL) + gy * VOXEL + gx;
        unsigned* dst = out + (size_t)cell * FDIM;
#if USE_ASYNC_LDS
        const float* srcf = wlds + q * FDIM;
#else
        const float* srcf = feat + (size_t)p * FDIM;
#endif
        // Each lane handles float4 at j*128 + lane*4 -> contiguous 512 B per
        // wave access: conflict-free LDS b128 reads, coalesced L2 atomics.
        for (int j = 0; j < 2; ++j) {
            int f0 = j * 128 + lane * 4;
            float4 v = *(const float4*)(srcf + f0);
            atomicMax(dst + f0 + 0, enc_ordered(v.x));
            atomicMax(dst + f0 + 1, enc_ordered(v.y));
            atomicMax(dst + f0 + 2, enc_ordered(v.z));
            atomicMax(dst + f0 + 3, enc_ordered(v.w));
        }
    }
}

__global__ __launch_bounds__(256) void decode_kernel(unsigned* __restrict__ data, int n4) {
    int i = blockIdx.x * blockDim.x + threadIdx.x;
    int stride = gridDim.x * blockDim.x;
    for (; i < n4; i += stride) {
        uint4 u = ((const uint4*)data)[i];
        float4 f;
        f.x = (u.x == 0u) ? 0.0f : __uint_as_float((u.x & 0x80000000u) ? (u.x ^ 0x80000000u) : ~u.x);
        f.y = (u.y == 0u) ? 0.0f : __uint_as_float((u.y & 0x80000000u) ? (u.y ^ 0x80000000u) : ~u.y);
        f.z = (u.z == 0u) ? 0.0f : __uint_as_float((u.z & 0x80000000u) ? (u.z ^ 0x80000000u) : ~u.z);
        f.w = (u.w == 0u) ? 0.0f : __uint_as_float((u.w & 0x80000000u) ? (u.w ^ 0x80000000u) : ~u.w);
        ((float4*)data)[i] = f;
    }
}

extern "C" void kernel_launch(void* const* d_in, const int* in_sizes, int n_in,
                              void* d_out, int out_size, void* d_ws, size_t ws_size,
                              hipStream_t stream) {
    const float* feat   = (const float*)d_in[0];   // [B, N, F] f32
    const float* coords = (const float*)d_in[1];   // [B, N, 2] f32
    unsigned* out = (unsigned*)d_out;              // reused as encoded u32 then decoded in place

    const int total_pts = in_sizes[1] / 2;                 // B*N
    const int cells     = out_size / FDIM;                 // B*V*V
    const int B         = cells / (VOXEL * VOXEL);
    const int pts_per_batch = (B > 0) ? (total_pts / B) : total_pts;

    const int n4 = out_size / 4;
    int initBlocks = (n4 + 255) / 256;
    if (initBlocks > 4096) initBlocks = 4096;              // grid-stride covers rest

    hipLaunchKernelGGL(init_zero_kernel, dim3(initBlocks), dim3(256), 0, stream,
                       (uint4*)d_out, n4);

    const int sBlocks = (total_pts + PTS_PER_BLOCK - 1) / PTS_PER_BLOCK;
    hipLaunchKernelGGL(scatter_max_kernel, dim3(sBlocks), dim3(256), 0, stream,
                       feat, coords, out, total_pts, pts_per_batch);

    hipLaunchKernelGGL(decode_kernel, dim3(initBlocks), dim3(256), 0, stream,
                       out, n4);
}